// A_P4CNN_31190052504419
// MI455X (gfx1250) — compile-verified
//
#include <hip/hip_runtime.h>

#define P4_B       2048
#define P4_NC      10
#define P4_NG      4
#define P4_EPS     2e-5f
#define BN_CHUNKS  128
#define CONV_WAVES 4
#define CONV_MT    4

typedef __attribute__((ext_vector_type(2))) float v2f;
typedef __attribute__((ext_vector_type(8))) float v8f;

static inline long hcdiv(long a, long b) { return (a + b - 1) / b; }

// ---------------------------------------------------------------------------
// Weight prep: rotate (and group-roll) conv weights into B-operand layout
// wbuf[r][K][n] with n in [0,16) (Cout padded 10->16 with zeros), K padded to K4.
// rot90 CCW source map applied r times: (si,sj) <- (sj, k-1-si).
// gg layers also roll group axis: source g = (gi - r) mod 4.
// ---------------------------------------------------------------------------
__global__ void p4_prep_weights(const float* __restrict__ w, float* __restrict__ wbuf,
                                int Cin, int Gin, int k, int Kreal, int K4, int doRoll) {
  int idx = blockIdx.x * blockDim.x + threadIdx.x;
  int total = 4 * K4 * 16;
  if (idx >= total) return;
  int n = idx & 15;
  int K = (idx >> 4) % K4;
  int r = idx / (K4 * 16);
  float val = 0.f;
  if (n < P4_NC && K < Kreal) {
    int kw = K % k; int t = K / k;
    int kh = t % k; t /= k;
    int gi = t % Gin; int ci = t / Gin;
    int si = kh, sj = kw;
    for (int it = 0; it < r; ++it) { int ni = sj; int nj = k - 1 - si; si = ni; sj = nj; }
    int gs = doRoll ? ((gi - r) & 3) : gi;
    long widx = ((((long)n * Cin + ci) * Gin + gs) * k + si) * k + sj;
    val = w[widx];
  }
  wbuf[idx] = val;
}

// ---------------------------------------------------------------------------
// CBAM channel attention: cA[b][ci] = sigmoid(MLP(avg) + MLP(max))
// act layout [B][Cin][Gin][H][W]; pooled over Gin*H*W. One block per b.
// ---------------------------------------------------------------------------
__global__ void p4_channel_att(const float* __restrict__ act,
                               const float* __restrict__ w1, const float* __restrict__ w2,
                               float* __restrict__ cA, int Cin, int hid, int GHW) {
  int b = blockIdx.x;
  int t = threadIdx.x;
  const float* p = act + (long)b * Cin * GHW;
  __shared__ float s_sum[256];
  __shared__ float s_max[256];
  __shared__ float avg_s[16], max_s[16];
  for (int ci = 0; ci < Cin; ++ci) {
    const float* q = p + (long)ci * GHW;
    float sm = 0.f, mx = -3.402823466e38f;
    for (int i = t; i < GHW; i += 256) { float v = q[i]; sm += v; mx = fmaxf(mx, v); }
    s_sum[t] = sm; s_max[t] = mx;
    __syncthreads();
    for (int s = 128; s > 0; s >>= 1) {
      if (t < s) { s_sum[t] += s_sum[t + s]; s_max[t] = fmaxf(s_max[t], s_max[t + s]); }
      __syncthreads();
    }
    if (t == 0) { avg_s[ci] = s_sum[0] / (float)GHW; max_s[ci] = s_max[0]; }
    __syncthreads();
  }
  if (t == 0) {
    float h_a[8], h_m[8];
    for (int h = 0; h < hid; ++h) {
      float sa = 0.f, sm2 = 0.f;
      for (int ci = 0; ci < Cin; ++ci) {
        float wv = w1[h * Cin + ci];
        sa += avg_s[ci] * wv; sm2 += max_s[ci] * wv;
      }
      h_a[h] = fmaxf(sa, 0.f); h_m[h] = fmaxf(sm2, 0.f);
    }
    for (int ci = 0; ci < Cin; ++ci) {
      float o = 0.f;
      for (int h = 0; h < hid; ++h) o += (h_a[h] + h_m[h]) * w2[ci * hid + h];
      cA[b * Cin + ci] = 1.f / (1.f + __expf(-o));
    }
  }
}

// stats[b][0][hw] = mean over (ci,gi), stats[b][1][hw] = max over (ci,gi)
__global__ void p4_spatial_stats(const float* __restrict__ act, float* __restrict__ stats,
                                 int B, int CG, int HW) {
  long idx = (long)blockIdx.x * blockDim.x + threadIdx.x;
  long total = (long)B * HW;
  if (idx >= total) return;
  int s = (int)(idx % HW);
  int b = (int)(idx / HW);
  const float* p = act + (long)b * CG * HW + s;
  float sm = 0.f, mx = -3.402823466e38f;
  for (int c = 0; c < CG; ++c) { float v = p[(long)c * HW]; sm += v; mx = fmaxf(mx, v); }
  stats[((long)b * 2 + 0) * HW + s] = sm / (float)CG;
  stats[((long)b * 2 + 1) * HW + s] = mx;
}

// sA[r][b][h][w] = sigmoid(conv7x7(stats, rot90^r(sw), pad=3)); r = blockIdx.y
__global__ void p4_spatial_att(const float* __restrict__ stats, const float* __restrict__ sw,
                               float* __restrict__ sA, int B, int H, int W) {
  __shared__ float swr[98];
  int r = blockIdx.y;
  for (int t = threadIdx.x; t < 98; t += blockDim.x) {
    int c = t / 49; int ij = t - c * 49;
    int i = ij / 7, j = ij % 7;
    int si = i, sj = j;
    for (int it = 0; it < r; ++it) { int ni = sj; int nj = 6 - si; si = ni; sj = nj; }
    swr[t] = sw[c * 49 + si * 7 + sj];
  }
  __syncthreads();
  long idx = (long)blockIdx.x * blockDim.x + threadIdx.x;
  long total = (long)B * H * W;
  if (idx >= total) return;
  int w = (int)(idx % W); long t = idx / W;
  int h = (int)(t % H);
  int b = (int)(t / H);
  float acc = 0.f;
  for (int c = 0; c < 2; ++c) {
    const float* sp = stats + ((long)b * 2 + c) * H * W;
    const float* kp = swr + c * 49;
    for (int i = 0; i < 7; ++i) {
      int ih = h + i - 3; if (ih < 0 || ih >= H) continue;
      for (int j = 0; j < 7; ++j) {
        int iw = w + j - 3; if (iw < 0 || iw >= W) continue;
        acc += sp[ih * W + iw] * kp[i * 7 + j];
      }
    }
  }
  sA[((long)r * B + b) * H * W + (long)h * W + w] = 1.f / (1.f + __expf(-acc));
}

// dst = src * cA[b][ci]  (src==dst allowed)
__global__ void p4_apply_cA(const float* __restrict__ src, float* __restrict__ dst,
                            const float* __restrict__ cA, int Cin, long GHW, long total) {
  long idx = (long)blockIdx.x * blockDim.x + threadIdx.x;
  if (idx >= total) return;
  int ci = (int)((idx / GHW) % Cin);
  int b  = (int)(idx / (GHW * Cin));
  dst[idx] = src[idx] * cA[b * Cin + ci];
}

// ---------------------------------------------------------------------------
// Implicit-GEMM conv via V_WMMA_F32_16X16X4_F32, 4 M-tiles per wave sharing B.
// A (16 x 4): lane<16 -> {A[m][K0+0], A[m][K0+1]}, lane>=16 -> K0+2/K0+3
// B (4 x 16): lane<16 -> rows K0+0/K0+1 at col n, lane>=16 -> rows K0+2/K0+3
// A[m][K] = xc[b][ci][gi][oh+kh][ow+kw] * sA[r][b][oh+kh][ow+kw]  (xc = act*cA)
// D 16x16: lane gives col n=lane&15; vgpr j -> row j + 8*(lane>=16)
// ---------------------------------------------------------------------------
__global__ void __launch_bounds__(CONV_WAVES * 32)
p4_conv_wmma(const float* __restrict__ xc, const float* __restrict__ sA,
             const float* __restrict__ wbuf, float* __restrict__ y,
             int B, int Cin, int Gin, int H, int W, int k,
             int OH, int OW, int Kreal, int Ksteps) {
  __shared__ int tabA[640];
  __shared__ int tabS[640];
  int K4 = Ksteps * 4;
  for (int K = threadIdx.x; K < K4; K += blockDim.x) {
    if (K < Kreal) {
      int kw = K % k; int t = K / k;
      int kh = t % k; t /= k;
      int gi = t % Gin; int ci = t / Gin;
      tabA[K] = ((ci * Gin + gi) * H + kh) * W + kw;
      tabS[K] = kh * W + kw;
    } else { tabA[K] = -1; tabS[K] = 0; }
  }
  __syncthreads();

  int lane = threadIdx.x & 31;
  int wv   = threadIdx.x >> 5;
  int r    = blockIdx.y;
  long Mtot = (long)B * OH * OW;
  int  mrow = lane & 15;
  int  hi   = lane >> 4;          // 0/1
  int  khalf2 = hi * 2;           // 0 or 2
  long tile0 = ((long)blockIdx.x * CONV_WAVES + wv) * CONV_MT;

  const float* wb = wbuf + (long)r * K4 * 16 + mrow;

  const float* xb[CONV_MT];
  const float* sb[CONV_MT];
  bool mval[CONV_MT];
#pragma unroll
  for (int ti = 0; ti < CONV_MT; ++ti) {
    long m = (tile0 + ti) * 16 + mrow;
    bool v = (m < Mtot);
    long mm = v ? m : 0;
    int ow = (int)(mm % OW); long t2 = mm / OW;
    int oh = (int)(t2 % OH);
    int b  = (int)(t2 / OH);
    xb[ti] = xc + (long)b * Cin * Gin * H * W + (long)oh * W + ow;
    sb[ti] = sA + ((long)r * B + b) * H * W + (long)oh * W + ow;
    mval[ti] = v;
  }

  v8f acc[CONV_MT] = {};
  for (int ks = 0; ks < Ksteps; ++ks) {
    int K0 = ks * 4 + khalf2;                 // always even -> pairable LDS reads
    int oa0 = tabA[K0], oa1 = tabA[K0 + 1];
    int os0 = tabS[K0], os1 = tabS[K0 + 1];
    v2f Bv; Bv.x = wb[(long)K0 * 16]; Bv.y = wb[(long)(K0 + 1) * 16];
#pragma unroll
    for (int ti = 0; ti < CONV_MT; ++ti) {
      float a0 = 0.f, a1 = 0.f;
      if (mval[ti]) {
        if (oa0 >= 0) a0 = xb[ti][oa0] * sb[ti][os0];
        if (oa1 >= 0) a1 = xb[ti][oa1] * sb[ti][os1];
      }
      v2f A; A.x = a0; A.y = a1;
      acc[ti] = __builtin_amdgcn_wmma_f32_16x16x4_f32(false, A, false, Bv,
                                                      (short)0, acc[ti], false, false);
    }
  }

  int n = mrow;                    // output channel
  if (n < P4_NC) {
    int mhi = hi * 8;
#pragma unroll
    for (int ti = 0; ti < CONV_MT; ++ti) {
      long mbase = (tile0 + ti) * 16 + mhi;
#pragma unroll
      for (int j = 0; j < 8; ++j) {
        long mo = mbase + j;
        if (mo < Mtot) {
          int ow2 = (int)(mo % OW); long tt = mo / OW;
          int oh2 = (int)(tt % OH);
          int b2  = (int)(tt / OH);
          y[(((long)b2 * P4_NC + n) * 4 + r) * OH * OW + (long)oh2 * OW + ow2] = acc[ti][j];
        }
      }
    }
  }
}

// ---------------------------------------------------------------------------
// Deterministic BatchNorm3d (training stats over B,G,H,W) + ReLU
// ---------------------------------------------------------------------------
__global__ void p4_bn_stats(const float* __restrict__ y, float* __restrict__ part,
                            int B, long S, int chunks) {
  int c = blockIdx.x;
  int chunk = blockIdx.y;
  long N = (long)B * S;
  long per = (N + chunks - 1) / chunks;
  long lo = (long)chunk * per;
  long hi = lo + per; if (hi > N) hi = N;
  float sm = 0.f, sq = 0.f;
  for (long i = lo + threadIdx.x; i < hi; i += blockDim.x) {
    long b = i / S; long s = i % S;
    float v = y[((long)b * P4_NC + c) * S + s];
    sm += v; sq += v * v;
  }
  __shared__ float s1[256], s2[256];
  s1[threadIdx.x] = sm; s2[threadIdx.x] = sq;
  __syncthreads();
  for (int s = 128; s > 0; s >>= 1) {
    if (threadIdx.x < s) { s1[threadIdx.x] += s1[threadIdx.x + s]; s2[threadIdx.x] += s2[threadIdx.x + s]; }
    __syncthreads();
  }
  if (threadIdx.x == 0) {
    part[c * chunks + chunk] = s1[0];
    part[(P4_NC + c) * chunks + chunk] = s2[0];
  }
}

__global__ void p4_bn_finalize(const float* __restrict__ part, float* __restrict__ ss,
                               const float* __restrict__ g, const float* __restrict__ bta,
                               int chunks, long Nred) {
  int c = threadIdx.x;
  if (c >= P4_NC) return;
  float sm = 0.f, sq = 0.f;
  for (int i = 0; i < chunks; ++i) { sm += part[c * chunks + i]; sq += part[(P4_NC + c) * chunks + i]; }
  float mean = sm / (float)Nred;
  float var  = sq / (float)Nred - mean * mean;
  float sc   = g[c] * rsqrtf(var + P4_EPS);
  ss[c] = sc;
  ss[P4_NC + c] = bta[c] - mean * sc;
}

__global__ void p4_bn_apply_relu(float* __restrict__ y, const float* __restrict__ ss,
                                 long S, long total) {
  long idx = (long)blockIdx.x * blockDim.x + threadIdx.x;
  if (idx >= total) return;
  int c = (int)((idx / S) % P4_NC);
  y[idx] = fmaxf(y[idx] * ss[c] + ss[P4_NC + c], 0.f);
}

__global__ void p4_maxpool2(const float* __restrict__ in, float* __restrict__ out,
                            long BCG, int H, int W) {
  int OHp = H / 2, OWp = W / 2;
  long idx = (long)blockIdx.x * blockDim.x + threadIdx.x;
  long total = BCG * OHp * OWp;
  if (idx >= total) return;
  int ow = (int)(idx % OWp); long t = idx / OWp;
  int oh = (int)(t % OHp);
  long bcg = t / OHp;
  const float* p = in + (bcg * H + oh * 2) * (long)W + ow * 2;
  out[idx] = fmaxf(fmaxf(p[0], p[1]), fmaxf(p[W], p[W + 1]));
}

__global__ void p4_final_max(const float* __restrict__ y7, float* __restrict__ out, int B) {
  int idx = blockIdx.x * blockDim.x + threadIdx.x;
  if (idx >= B * P4_NC) return;
  const float* p = y7 + (long)idx * 4;
  out[idx] = fmaxf(fmaxf(p[0], p[1]), fmaxf(p[2], p[3]));
}

// ---------------------------------------------------------------------------
extern "C" void kernel_launch(void* const* d_in, const int* in_sizes, int n_in,
                              void* d_out, int out_size, void* d_ws, size_t ws_size,
                              hipStream_t stream) {
  (void)in_sizes; (void)n_in; (void)out_size; (void)ws_size;
  const int B = P4_B;
  const float* x = (const float*)d_in[0];

  // params flattened in dict insertion order: x, c1{w,cw1,cw2,sw}, c2..c7{...}, bn1..bn6{g,b}
  const float *W_[8], *W1_[8], *W2_[8], *SW_[8];
  W_[1]  = (const float*)d_in[1]; W1_[1] = (const float*)d_in[2];
  W2_[1] = (const float*)d_in[3]; SW_[1] = (const float*)d_in[4];
  for (int i = 2; i <= 7; ++i) {
    int base = 5 + (i - 2) * 4;
    W_[i]  = (const float*)d_in[base];     W1_[i] = (const float*)d_in[base + 1];
    W2_[i] = (const float*)d_in[base + 2]; SW_[i] = (const float*)d_in[base + 3];
  }
  const float *BNg[7], *BNb[7];
  for (int i = 1; i <= 6; ++i) {
    BNg[i] = (const float*)d_in[29 + (i - 1) * 2];
    BNb[i] = (const float*)d_in[30 + (i - 1) * 2];
  }

  // workspace layout (floats)
  float* ws = (float*)d_ws;
  size_t off = 0;
  const size_t BIG = (size_t)B * P4_NC * 4 * 26 * 26;     // 55,377,920 floats
  float* bufA  = ws + off; off += BIG;
  float* bufB  = ws + off; off += BIG;
  float* sAb   = ws + off; off += (size_t)4 * B * 28 * 28;
  float* stats = ws + off; off += (size_t)B * 2 * 28 * 28;
  float* cAb   = ws + off; off += (size_t)B * P4_NC;
  float* wbuf  = ws + off; off += (size_t)4 * 640 * 16;
  float* part  = ws + off; off += (size_t)2 * P4_NC * BN_CHUNKS;
  float* ss    = ws + off; off += (size_t)2 * P4_NC;

  auto attconv = [&](const float* act, float* xcdst, float* yout,
                     int Cin, int Gin, int H, int k, int hid, int li) {
    int HW = H * H, GHW = Gin * HW, CG = Cin * Gin;
    int Kreal = Cin * Gin * k * k;
    int Ksteps = (Kreal + 3) / 4, K4 = Ksteps * 4;
    int OH = H - k + 1;
    {
      long tot = 4L * K4 * 16;
      p4_prep_weights<<<(unsigned)hcdiv(tot, 256), 256, 0, stream>>>(
          W_[li], wbuf, Cin, Gin, k, Kreal, K4, (Gin == 4) ? 1 : 0);
    }
    p4_channel_att<<<B, 256, 0, stream>>>(act, W1_[li], W2_[li], cAb, Cin, hid, GHW);
    {
      long tot = (long)B * HW;
      p4_spatial_stats<<<(unsigned)hcdiv(tot, 256), 256, 0, stream>>>(act, stats, B, CG, HW);
    }
    {
      long tot = (long)B * HW;
      dim3 g((unsigned)hcdiv(tot, 256), 4);
      p4_spatial_att<<<g, 256, 0, stream>>>(stats, SW_[li], sAb, B, H, H);
    }
    {
      long tot = (long)B * Cin * GHW;
      p4_apply_cA<<<(unsigned)hcdiv(tot, 256), 256, 0, stream>>>(act, xcdst, cAb, Cin, (long)GHW, tot);
    }
    {
      long Mtot = (long)B * OH * OH;
      long tiles = hcdiv(Mtot, 16);
      dim3 g((unsigned)hcdiv(tiles, CONV_WAVES * CONV_MT), 4);
      p4_conv_wmma<<<g, CONV_WAVES * 32, 0, stream>>>(
          xcdst, sAb, wbuf, yout, B, Cin, Gin, H, H, k, OH, OH, Kreal, Ksteps);
    }
  };

  auto bnrelu = [&](float* y, int OH, int li) {
    long S = 4L * OH * OH;
    long Nred = (long)B * S;
    long tot = (long)B * P4_NC * S;
    p4_bn_stats<<<dim3(P4_NC, BN_CHUNKS), 256, 0, stream>>>(y, part, B, S, BN_CHUNKS);
    p4_bn_finalize<<<1, 32, 0, stream>>>(part, ss, BNg[li], BNb[li], BN_CHUNKS, Nred);
    p4_bn_apply_relu<<<(unsigned)hcdiv(tot, 256), 256, 0, stream>>>(y, ss, S, tot);
  };

  // L1 (lifting R^2 -> P4): x[2048,1,28,28] -> bufA[2048,10,4,26,26]; xc staged in bufB
  attconv(x, bufB, bufA, /*Cin*/1, /*Gin*/1, 28, 3, /*hid*/1, 1);
  bnrelu(bufA, 26, 1);
  // L2: bufA(26) -> bufB(24)
  attconv(bufA, bufA, bufB, 10, 4, 26, 3, 5, 2);
  bnrelu(bufB, 24, 2);
  // maxpool 24 -> 12
  {
    long BCG = (long)B * P4_NC * 4;
    long tot = BCG * 12 * 12;
    p4_maxpool2<<<(unsigned)hcdiv(tot, 256), 256, 0, stream>>>(bufB, bufA, BCG, 24, 24);
  }
  // L3..L6 (ping-pong)
  attconv(bufA, bufA, bufB, 10, 4, 12, 3, 5, 3); bnrelu(bufB, 10, 3);
  attconv(bufB, bufB, bufA, 10, 4, 10, 3, 5, 4); bnrelu(bufA, 8, 4);
  attconv(bufA, bufA, bufB, 10, 4, 8, 3, 5, 5);  bnrelu(bufB, 6, 5);
  attconv(bufB, bufB, bufA, 10, 4, 6, 3, 5, 6);  bnrelu(bufA, 4, 6);
  // L7: 4x4 conv -> [B,10,4,1,1], no BN
  attconv(bufA, bufA, bufB, 10, 4, 4, 4, 5, 7);
  // orientation max -> [B,10]
  p4_final_max<<<(unsigned)hcdiv((long)B * P4_NC, 256), 256, 0, stream>>>(bufB, (float*)d_out, B);
}